// EdgeClassifierGNN_31619549233491
// MI455X (gfx1250) — compile-verified
//
#include <hip/hip_runtime.h>
#include <hip/hip_bf16.h>

#define N_NODES 50000
#define IN_CH   128
#define HID     128
#define OUT_CH  2
#define N_EDGES 800000

typedef __attribute__((ext_vector_type(2))) float v2f;
typedef __attribute__((ext_vector_type(8))) float v8f;

// ---------------------------------------------------------------------------
// Scatter: one wave (32 lanes) per edge. Lane l handles features [4l,4l+4).
// agg[dst] += feat[src]; optionally count degree (layer 1 only).
// ---------------------------------------------------------------------------
__global__ __launch_bounds__(256) void gnn_scatter_kernel(
    const float* __restrict__ feat, const int* __restrict__ ei,
    float* __restrict__ agg, float* __restrict__ cnt, int with_count) {
  const int wave = (blockIdx.x * blockDim.x + threadIdx.x) >> 5;
  const int lane = threadIdx.x & 31;
  if (wave >= N_EDGES) return;
  const int s = ei[wave];
  const int d = ei[N_EDGES + wave];
  const float4 v = *(const float4*)(feat + (size_t)s * HID + lane * 4);
  float* p = agg + (size_t)d * HID + lane * 4;
  atomicAdd(p + 0, v.x);
  atomicAdd(p + 1, v.y);
  atomicAdd(p + 2, v.z);
  atomicAdd(p + 3, v.w);
  if (with_count && lane == 0) atomicAdd(cnt + d, 1.0f);
}

// ---------------------------------------------------------------------------
// SAGEConv dense part via fp32 WMMA (V_WMMA_F32_16X16X4_F32):
//   hout = relu( (agg/max(cnt,1)) @ Wl^T + bl + hin @ Wr^T )
// Block = 256 threads = 8 waves; one 16-node row tile per block, each wave
// produces a 16x16 output tile (16 of the 128 output channels).
// ---------------------------------------------------------------------------
__global__ __launch_bounds__(256) void gnn_sage_wmma_kernel(
    const float* __restrict__ agg, const float* __restrict__ cnt,
    const float* __restrict__ hin,
    const float* __restrict__ Wl, const float* __restrict__ Wr,
    const float* __restrict__ bl, float* __restrict__ hout) {
  __shared__ float sm_mean[16][132];  // pitch 132 -> conflict-free b64 reads
  __shared__ float sm_x[16][132];
  __shared__ float sinv[16];

  const int tid  = threadIdx.x;
  const int base = blockIdx.x * 16;  // 50000 = 3125 * 16, no tail

  if (tid < 16) {
    const float c = cnt[base + tid];
    sinv[tid] = 1.0f / fmaxf(c, 1.0f);
  }
  __syncthreads();

  for (int idx = tid; idx < 16 * HID; idx += 256) {
    const int r = idx >> 7;
    const int k = idx & 127;
    const size_t off = (size_t)(base + r) * HID + k;
    sm_mean[r][k] = agg[off] * sinv[r];
    sm_x[r][k]    = hin[off];
  }
  __syncthreads();

  const int lane = tid & 31;
  const int wave = tid >> 5;
  const int r    = lane & 15;        // A-row / B-col within tile
  const int kk   = (lane >> 4) << 1; // K sub-offset per ISA 16x4 f32 layout
  const int col  = wave * 16 + r;    // output channel this lane feeds

  const float* wlrow = Wl + (size_t)col * HID;  // B[k][col] = W[col][k]
  const float* wrrow = Wr + (size_t)col * HID;

  v8f acc = {};
#pragma unroll
  for (int k0 = 0; k0 < HID; k0 += 4) {
    const v2f a = *(const v2f*)&sm_mean[r][k0 + kk];
    const v2f b = *(const v2f*)&wlrow[k0 + kk];
    acc = __builtin_amdgcn_wmma_f32_16x16x4_f32(false, a, false, b,
                                                (short)0, acc, false, false);
  }
#pragma unroll
  for (int k0 = 0; k0 < HID; k0 += 4) {
    const v2f a = *(const v2f*)&sm_x[r][k0 + kk];
    const v2f b = *(const v2f*)&wrrow[k0 + kk];
    acc = __builtin_amdgcn_wmma_f32_16x16x4_f32(false, a, false, b,
                                                (short)0, acc, false, false);
  }

  const float bias  = bl[col];
  const int   rbase = (lane >> 4) * 8;  // C/D layout: lanes>=16 hold M=v+8
#pragma unroll
  for (int v = 0; v < 8; ++v) {
    float val = acc[v] + bias;
    val = fmaxf(val, 0.0f);
    hout[(size_t)(base + rbase + v) * HID + col] = val;
  }
}

// ---------------------------------------------------------------------------
// Edge classifier: one wave per edge.
// out[e] = [h[src], h[dst]] @ Wc^T + bc   (Wc is [2, 256])
// ---------------------------------------------------------------------------
__global__ __launch_bounds__(256) void gnn_classify_kernel(
    const float* __restrict__ h, const int* __restrict__ ei,
    const float* __restrict__ Wc, const float* __restrict__ bc,
    float* __restrict__ out) {
  const int wave = (blockIdx.x * blockDim.x + threadIdx.x) >> 5;
  const int lane = threadIdx.x & 31;
  if (wave >= N_EDGES) return;
  const int s = ei[wave];
  const int d = ei[N_EDGES + wave];

  const float4 hs = *(const float4*)(h + (size_t)s * HID + lane * 4);
  const float4 hd = *(const float4*)(h + (size_t)d * HID + lane * 4);
  const float4 w0s = *(const float4*)(Wc + lane * 4);
  const float4 w0d = *(const float4*)(Wc + HID + lane * 4);
  const float4 w1s = *(const float4*)(Wc + 2 * HID + lane * 4);
  const float4 w1d = *(const float4*)(Wc + 3 * HID + lane * 4);

  float a0 = hs.x * w0s.x + hs.y * w0s.y + hs.z * w0s.z + hs.w * w0s.w +
             hd.x * w0d.x + hd.y * w0d.y + hd.z * w0d.z + hd.w * w0d.w;
  float a1 = hs.x * w1s.x + hs.y * w1s.y + hs.z * w1s.z + hs.w * w1s.w +
             hd.x * w1d.x + hd.y * w1d.y + hd.z * w1d.z + hd.w * w1d.w;

#pragma unroll
  for (int off = 16; off > 0; off >>= 1) {
    a0 += __shfl_down(a0, off, 32);
    a1 += __shfl_down(a1, off, 32);
  }
  if (lane == 0) {
    out[(size_t)wave * OUT_CH + 0] = a0 + bc[0];
    out[(size_t)wave * OUT_CH + 1] = a1 + bc[1];
  }
}

// ---------------------------------------------------------------------------
extern "C" void kernel_launch(void* const* d_in, const int* in_sizes, int n_in,
                              void* d_out, int out_size, void* d_ws, size_t ws_size,
                              hipStream_t stream) {
  (void)in_sizes; (void)n_in; (void)out_size; (void)ws_size;

  const float* x   = (const float*)d_in[0];
  const int*   ei  = (const int*)d_in[1];  // [2, E]: row0 = src, row1 = dst
  const float* W1l = (const float*)d_in[2];
  const float* b1l = (const float*)d_in[3];
  const float* W1r = (const float*)d_in[4];
  const float* W2l = (const float*)d_in[5];
  const float* b2l = (const float*)d_in[6];
  const float* W2r = (const float*)d_in[7];
  const float* Wc  = (const float*)d_in[8];
  const float* bc  = (const float*)d_in[9];
  float* out = (float*)d_out;

  // Workspace layout: [cnt | agg | h1 | h2]
  char* ws = (char*)d_ws;
  const size_t cnt_bytes  = ((size_t)N_NODES * sizeof(float) + 511) & ~(size_t)511;
  const size_t feat_bytes = (size_t)N_NODES * HID * sizeof(float);
  float* cnt = (float*)(ws);
  float* agg = (float*)(ws + cnt_bytes);
  float* h1  = (float*)(ws + cnt_bytes + feat_bytes);
  float* h2  = (float*)(ws + cnt_bytes + 2 * feat_bytes);

  const int edge_blocks = (N_EDGES * 32 + 255) / 256;  // one wave per edge
  const int node_blocks = N_NODES / 16;                // 3125, exact

  // ---- Layer 1 ----
  hipMemsetAsync(ws, 0, cnt_bytes + feat_bytes, stream);  // cnt + agg
  gnn_scatter_kernel<<<edge_blocks, 256, 0, stream>>>(x, ei, agg, cnt, 1);
  gnn_sage_wmma_kernel<<<node_blocks, 256, 0, stream>>>(agg, cnt, x, W1l, W1r,
                                                        b1l, h1);
  // ---- Layer 2 ----
  hipMemsetAsync(agg, 0, feat_bytes, stream);
  gnn_scatter_kernel<<<edge_blocks, 256, 0, stream>>>(h1, ei, agg, cnt, 0);
  gnn_sage_wmma_kernel<<<node_blocks, 256, 0, stream>>>(agg, cnt, h1, W2l, W2r,
                                                        b2l, h2);
  // ---- Edge classifier ----
  gnn_classify_kernel<<<edge_blocks, 256, 0, stream>>>(h2, ei, Wc, bc, out);
}